// GCN_Model_74337293959431
// MI455X (gfx1250) — compile-verified
//
#include <hip/hip_runtime.h>

// GCN on MI455X (gfx1250): fp32 WMMA GEMMs, TDM tensor_load_to_lds staging,
// hardware global_atomic_add_f32 scatter (L2-resident working set).

typedef float v2f __attribute__((ext_vector_type(2)));
typedef float v8f __attribute__((ext_vector_type(8)));
typedef unsigned int u32x4 __attribute__((ext_vector_type(4)));
typedef int i32x4 __attribute__((ext_vector_type(4)));
typedef int i32x8 __attribute__((ext_vector_type(8)));

#define NN 100000
#define H  128

#if defined(__has_builtin)
#if __has_builtin(__builtin_amdgcn_tensor_load_to_lds) && __has_builtin(__builtin_amdgcn_s_wait_tensorcnt)
#define USE_TDM 1
#endif
#endif
#ifndef USE_TDM
#define USE_TDM 0
#endif

__device__ __forceinline__ void atomAdd(float* p, float v) {
#if defined(__gfx1250__)
  // Pin the hardware f32 atomic at device scope (lands in L2).
  asm volatile("global_atomic_add_f32 %0, %1, off scope:SCOPE_DEV"
               :: "v"(p), "v"(v) : "memory");
#else
  __hip_atomic_fetch_add(p, v, __ATOMIC_RELAXED, __HIP_MEMORY_SCOPE_AGENT);
#endif
}

#if USE_TDM
// Issue a TDM 2D tile load: tile_h rows x tile_w dwords (row stride `stride`
// dwords in memory), zero-filling rows past tensor_h, writing LDS rows of
// (tile_w + 4*pad_en) dwords. pad_iv/pad_amt are the ISA-encoded fields.
__device__ __forceinline__ void tdm_load_2d(unsigned lds_off, const void* gptr,
                                            unsigned tile_w, unsigned tile_h,
                                            unsigned stride, unsigned tensor_h,
                                            unsigned pad_en, unsigned pad_iv,
                                            unsigned pad_amt) {
  unsigned long long ga = (unsigned long long)gptr;
  u32x4 g0;
  g0[0] = 1u;                                       // count=1, user descriptor
  g0[1] = lds_off;                                  // LDS byte address
  g0[2] = (unsigned)ga;                             // global_addr[31:0]
  g0[3] = (unsigned)((ga >> 32) & 0x01FFFFFFu) | (2u << 30);  // addr[56:32], type=2

  i32x8 g1;
  g1[0] = (int)((2u << 16) | (pad_en << 20) | (pad_iv << 22) | (pad_amt << 25)); // data_size=4B
  g1[1] = (int)((tile_w & 0xFFFFu) << 16);          // tensor_dim0[15:0] (== tile_w)
  g1[2] = (int)((tile_w >> 16) | ((tensor_h & 0xFFFFu) << 16));  // dim0 hi | dim1 lo
  g1[3] = (int)((tensor_h >> 16) | (tile_w << 16)); // dim1 hi | tile_dim0
  g1[4] = (int)(tile_h & 0xFFFFu);                  // tile_dim1; tile_dim2=0 (2D)
  g1[5] = (int)stride;                              // tensor_dim0_stride[31:0]
  g1[6] = 0;                                        // stride hi | dim1_stride lo
  g1[7] = 0;
  i32x4 z4 = {0, 0, 0, 0};
  i32x8 z8 = {0, 0, 0, 0, 0, 0, 0, 0};
  // 6-arg toolchain form: (g0, g1, g2, g3, extra, cpol)
  __builtin_amdgcn_tensor_load_to_lds(g0, g1, z4, z4, z8, 0);
}
#endif

// ---------------- degree / norm precompute ----------------
__global__ __launch_bounds__(256) void k_deg_init(float* __restrict__ deg, int n) {
  int i = blockIdx.x * 256 + threadIdx.x;
  if (i < n) deg[i] = 1.0f;  // self-loop contributes 1 to every node
}

__global__ __launch_bounds__(256) void k_deg_accum(const long long* __restrict__ ei,
                                                   float* __restrict__ deg, int E) {
  int i = blockIdx.x * 256 + threadIdx.x;
  if (i < E) atomAdd(&deg[(int)ei[(size_t)E + i]], 1.0f);
}

__global__ __launch_bounds__(256) void k_dinv(float* __restrict__ deg, int n) {
  int i = blockIdx.x * 256 + threadIdx.x;
  if (i < n) { float d = deg[i]; deg[i] = (d > 0.0f) ? rsqrtf(d) : 0.0f; }
}

__global__ __launch_bounds__(256) void k_norm(const long long* __restrict__ ei,
                                              const float* __restrict__ dinv,
                                              float* __restrict__ nrm, int E) {
  int i = blockIdx.x * 256 + threadIdx.x;
  if (i < E) nrm[i] = dinv[(int)ei[i]] * dinv[(int)ei[(size_t)E + i]];
}

// ---------------- fp32 WMMA GEMM: C[M x NCOLS] = A[M x 128] * W[128 x NCOLS] ----------------
// Block: 8 waves. NCOLS=128 -> 4 row-waves x 2 col-waves, 64-row tile.
//        NCOLS=64  -> 8 row-waves x 1 col-wave, 128-row tile.
// Each wave: 16 rows x 64 cols = 4 accumulators of v_wmma_f32_16x16x4_f32.
template <int NCOLS>
__global__ __launch_bounds__(256)
void k_gemm_wmma(const float* __restrict__ A, const float* __restrict__ W,
                 const float* __restrict__ bias, float* __restrict__ C,
                 int M, int addBias)
{
  constexpr int COLG  = NCOLS / 64;   // waves along N
  constexpr int ROWG  = 8 / COLG;     // waves along M
  constexpr int BROWS = ROWG * 16;
  constexpr int KS    = 32;           // K slab
  constexpr int K     = 128;
  // TDM pad_interval encoding: value v pads every 2^(v+1) dwords.
  constexpr unsigned PAD_IV_A = 4;                      // every 32 dwords
  constexpr unsigned PAD_IV_W = (NCOLS == 128) ? 6 : 5; // every NCOLS dwords
  constexpr unsigned PAD_AMT  = 3;                      // pad 4 dwords

  __shared__ float As[BROWS][KS + 4];     // pad 4 floats -> rows stay 16B aligned
  __shared__ float Ws[KS][NCOLS + 4];

  const int tid  = threadIdx.x;
  const int wave = tid >> 5;
  const int lane = tid & 31;
  const int wr   = wave / COLG;
  const int wc   = wave % COLG;
  const int ml   = lane & 15;         // M (A/C) or N (B) index within tile
  const int kh   = lane >> 4;         // lane half selects K pair / C row half
  const int rowBase = blockIdx.x * BROWS;

  v8f acc[4];
#pragma unroll
  for (int t = 0; t < 4; ++t)
#pragma unroll
    for (int v = 0; v < 8; ++v) acc[t][v] = 0.0f;

  for (int ks = 0; ks < K; ks += KS) {
#if USE_TDM
    if (wave == 0) {
      unsigned rowsLeft = (unsigned)(M - rowBase);   // TDM zero-fills rows past this
      tdm_load_2d((unsigned)(size_t)&As[0][0], A + (size_t)rowBase * K + ks,
                  KS, BROWS, K, rowsLeft, 1u, PAD_IV_A, PAD_AMT);
      tdm_load_2d((unsigned)(size_t)&Ws[0][0], W + (size_t)ks * NCOLS,
                  NCOLS, KS, NCOLS, KS, 1u, PAD_IV_W, PAD_AMT);
      __builtin_amdgcn_s_wait_tensorcnt(0);
    }
    __syncthreads();
#else
    // stage A slab: BROWS x 32 floats (clamp OOB rows; stores are guarded later)
#pragma unroll
    for (int i = tid; i < BROWS * (KS / 4); i += 256) {
      int r  = i >> 3;
      int c4 = i & 7;
      int gr = rowBase + r; if (gr >= M) gr = M - 1;
      float4 v = *(const float4*)(A + (size_t)gr * K + ks + c4 * 4);
      *(float4*)(&As[r][c4 * 4]) = v;
    }
    // stage W slab: 32 x NCOLS floats
#pragma unroll
    for (int i = tid; i < KS * (NCOLS / 4); i += 256) {
      int r  = i / (NCOLS / 4);
      int c4 = i % (NCOLS / 4);
      float4 v = *(const float4*)(W + (size_t)(ks + r) * NCOLS + c4 * 4);
      *(float4*)(&Ws[r][c4 * 4]) = v;
    }
    __syncthreads();
#endif

#pragma unroll
    for (int k4 = 0; k4 < KS / 4; ++k4) {
      const int kb = k4 * 4 + 2 * kh;   // A: lanes0-15 K={kb,kb+1}, lanes16-31 +2
      v2f a;
      a.x = As[wr * 16 + ml][kb];
      a.y = As[wr * 16 + ml][kb + 1];
#pragma unroll
      for (int t = 0; t < 4; ++t) {
        v2f b;
        b.x = Ws[kb][wc * 64 + t * 16 + ml];
        b.y = Ws[kb + 1][wc * 64 + t * 16 + ml];
        acc[t] = __builtin_amdgcn_wmma_f32_16x16x4_f32(
            false, a, false, b, (short)0, acc[t], false, false);
      }
    }
    __syncthreads();
  }

  // C layout: VGPR v -> M=v (lanes 0-15) / M=v+8 (lanes 16-31), N=lane%16
#pragma unroll
  for (int t = 0; t < 4; ++t) {
#pragma unroll
    for (int v = 0; v < 8; ++v) {
      int r = rowBase + wr * 16 + v + 8 * kh;
      int c = wc * 64 + t * 16 + ml;
      if (r < M) {
        float val = acc[t][v];
        if (addBias) val += bias[c];
        C[(size_t)r * NCOLS + c] = val;
      }
    }
  }
}

// ---------------- aggregation (scatter-add over edges) ----------------
// out[n,:] = dinv[n]^2 * h[n,:]   (self-loop term; also serves as zero-init)
__global__ __launch_bounds__(256)
void k_agg_init(const float* __restrict__ h, const float* __restrict__ dinv,
                float* __restrict__ out, int n) {
  int i = blockIdx.x * 256 + threadIdx.x;   // one float4 per thread, 32 per node
  if (i >= n * 32) return;
  int node = i >> 5, sub = i & 31;
  float s = dinv[node]; s = s * s;
  float4 v = *(const float4*)(h + (size_t)node * H + sub * 4);
  v.x *= s; v.y *= s; v.z *= s; v.w *= s;
  *(float4*)(out + (size_t)node * H + sub * 4) = v;
}

// out[col,:] += norm[e] * h[row,:]  -- one wave per edge, float4 per lane
__global__ __launch_bounds__(256)
void k_agg_edges(const long long* __restrict__ ei, const float* __restrict__ nrm,
                 const float* __restrict__ h, float* __restrict__ out, int E) {
  int gid = blockIdx.x * 256 + threadIdx.x;
  int e = gid >> 5, sub = gid & 31;
  if (e >= E) return;
  int r = (int)ei[e];
  int c = (int)ei[(size_t)E + e];
  float nm = nrm[e];
  float4 v = *(const float4*)(h + (size_t)r * H + sub * 4);
  float* dst = out + (size_t)c * H + sub * 4;
  atomAdd(dst + 0, v.x * nm);
  atomAdd(dst + 1, v.y * nm);
  atomAdd(dst + 2, v.z * nm);
  atomAdd(dst + 3, v.w * nm);
}

__global__ __launch_bounds__(256)
void k_bias_relu(float* __restrict__ out, const float* __restrict__ bias, int n) {
  int i = blockIdx.x * 256 + threadIdx.x;   // float4 granularity over [n,128]
  if (i >= n * 32) return;
  int sub = i & 31;
  float4 v = *(float4*)(out + (size_t)i * 4);
  float4 b = *(const float4*)(bias + sub * 4);
  v.x = fmaxf(v.x + b.x, 0.0f);
  v.y = fmaxf(v.y + b.y, 0.0f);
  v.z = fmaxf(v.z + b.z, 0.0f);
  v.w = fmaxf(v.w + b.w, 0.0f);
  *(float4*)(out + (size_t)i * 4) = v;
}

// ---------------- host-side launch ----------------
extern "C" void kernel_launch(void* const* d_in, const int* in_sizes, int n_in,
                              void* d_out, int out_size, void* d_ws, size_t ws_size,
                              hipStream_t stream) {
  (void)n_in; (void)out_size; (void)ws_size;
  const float*     x  = (const float*)d_in[0];
  const long long* ei = (const long long*)d_in[1];   // int64 [2, E] flat
  const float* W1 = (const float*)d_in[2];
  const float* b1 = (const float*)d_in[3];
  const float* W2 = (const float*)d_in[4];
  const float* b2 = (const float*)d_in[5];
  const float* W3 = (const float*)d_in[6];
  const float* b3 = (const float*)d_in[7];
  const float* Wo = (const float*)d_in[8];
  const float* bo = (const float*)d_in[9];
  float* out = (float*)d_out;

  const int N = NN;
  const int E = in_sizes[1] / 2;

  // workspace carve-out
  char* ws = (char*)d_ws;
  size_t off = 0;
  auto take = [&](size_t nbytes) -> float* {
    float* p = (float*)(ws + off);
    off = (off + nbytes + 255) & ~(size_t)255;
    return p;
  };
  float* dinv = take((size_t)N * sizeof(float));       // deg -> dinv in place
  float* nrm  = take((size_t)E * sizeof(float));
  float* G    = take((size_t)N * H * sizeof(float));   // GEMM output
  float* Ag   = take((size_t)N * H * sizeof(float));   // aggregated output

  const int TB = 256;
  dim3 blk(TB);
  int gN   = (N + TB - 1) / TB;
  int gE   = (E + TB - 1) / TB;
  int gNH  = (N * 32 + TB - 1) / TB;
  int gEH  = (int)(((long long)E * 32 + TB - 1) / TB);
  int g128 = (N + 63) / 64;
  int g64  = (N + 127) / 128;

  k_deg_init <<<gN, blk, 0, stream>>>(dinv, N);
  k_deg_accum<<<gE, blk, 0, stream>>>(ei, dinv, E);
  k_dinv     <<<gN, blk, 0, stream>>>(dinv, N);
  k_norm     <<<gE, blk, 0, stream>>>(ei, dinv, nrm, E);

  const float* in = x;
  const float* Wl[3] = {W1, W2, W3};
  const float* bl[3] = {b1, b2, b3};
  for (int l = 0; l < 3; ++l) {
    k_gemm_wmma<128><<<g128, blk, 0, stream>>>(in, Wl[l], nullptr, G, N, 0);
    k_agg_init      <<<gNH, blk, 0, stream>>>(G, dinv, Ag, N);
    k_agg_edges     <<<gEH, blk, 0, stream>>>(ei, nrm, G, Ag, E);
    k_bias_relu     <<<gNH, blk, 0, stream>>>(Ag, bl[l], N);
    in = Ag;
  }
  // final projection with bias, no relu, straight into d_out [N, 64]
  k_gemm_wmma<64><<<g64, blk, 0, stream>>>(Ag, Wo, bo, out, N, 1);
}